// MultiHeadGATLayerWithEdgeFeatures_8504035246421
// MI455X (gfx1250) — compile-verified
//
#include <hip/hip_runtime.h>
#include <math.h>

#define HEADS   4
#define OUT_F   32
#define EDGE_F  16
#define IN_F    128
#define NNODES  20000
#define NEDGES  640000
#define ALPHA   0.2f
#define A_ROW   (2*OUT_F + EDGE_F)   // 80
#define OUTW    (HEADS*OUT_F)        // 128

typedef __attribute__((ext_vector_type(2))) float v2f;
typedef __attribute__((ext_vector_type(8))) float v8f;

// ---------------------------------------------------------------------------
// Kernel 1: Wh = h @ W   (20000x128 @ 128x128), fp32 WMMA 16x16x4.
// One wave32 per 16x16 output tile; K looped in steps of 4.
// A layout (16x4 f32): lane<16 -> row=lane, K=k0..k0+1 ; lane>=16 -> K=k0+2..3
// B layout (4x16 f32): mirrored (col=lane&15, K split by lane>=16)
// C/D layout: VGPR v -> row = v + (lane>=16 ? 8 : 0), col = lane&15
// ---------------------------------------------------------------------------
__global__ __launch_bounds__(256) void gat_gemm_wmma(const float* __restrict__ h,
                                                     const float* __restrict__ W,
                                                     float* __restrict__ Wh) {
  const int lane = threadIdx.x & 31;
  const int wid  = threadIdx.x >> 5;
  const int tile = blockIdx.x * 8 + wid;          // 8 waves/block
  const int rowTile = tile >> 3;                  // 128/16 = 8 col tiles
  const int colTile = tile & 7;
  if (rowTile >= NNODES / 16) return;
  const int row0 = rowTile * 16;
  const int col0 = colTile * 16;

  const int mrow  = lane & 15;
  const int khalf = (lane >> 4) << 1;             // 0 or 2

  v8f c = {};
#pragma unroll
  for (int k0 = 0; k0 < IN_F; k0 += 4) {
    const int ka = k0 + khalf;
    v2f a, b;
    a.x = h[(row0 + mrow) * IN_F + ka];
    a.y = h[(row0 + mrow) * IN_F + ka + 1];
    b.x = W[ ka      * OUTW + col0 + mrow];
    b.y = W[(ka + 1) * OUTW + col0 + mrow];
    // 8 args: (neg_a, A, neg_b, B, c_mod, C, reuse_a, reuse_b)
    c = __builtin_amdgcn_wmma_f32_16x16x4_f32(false, a, false, b,
                                              (short)0, c, false, false);
  }

  const int rbase = row0 + ((lane >> 4) << 3);
  const int col   = col0 + mrow;
#pragma unroll
  for (int v = 0; v < 8; ++v)
    Wh[(rbase + v) * OUTW + col] = c[v];
}

// ---------------------------------------------------------------------------
// Kernel 2: per-node attention partial scores
//   s_src[n,h] = Wh[n,h,:] . a[h, 0:32]
//   s_dst[n,h] = Wh[n,h,:] . a[h, 32:64]
// ---------------------------------------------------------------------------
__global__ void gat_node_scores(const float* __restrict__ Wh,
                                const float* __restrict__ a,
                                float* __restrict__ s_src,
                                float* __restrict__ s_dst) {
  const int i = blockIdx.x * blockDim.x + threadIdx.x;   // [0, N*HEADS)
  if (i >= NNODES * HEADS) return;
  const int n  = i >> 2;
  const int hh = i & 3;
  const float* wrow = Wh + n * OUTW + hh * OUT_F;
  const float* asrc = a + hh * A_ROW;
  const float* adst = asrc + OUT_F;
  float ss = 0.f, sd = 0.f;
#pragma unroll
  for (int d = 0; d < OUT_F; ++d) {
    const float w = wrow[d];
    ss += w * asrc[d];
    sd += w * adst[d];
  }
  s_src[i] = ss;
  s_dst[i] = sd;
}

// ---------------------------------------------------------------------------
// Kernel 3: edge stage — one wave32 per edge.
// lanes 0..3 compute per-head logits (+ leaky relu); softmax over the 4 heads
// is broadcast by wave shuffles; all 32 lanes then do 4 coalesced
// global_atomic_add_f32 groups into out[src] (L2-resident, 10 MB).
// ---------------------------------------------------------------------------
__global__ __launch_bounds__(256) void gat_edges(const int*   __restrict__ ei,
                                                 const float* __restrict__ ea,
                                                 const float* __restrict__ a,
                                                 const float* __restrict__ Wh,
                                                 const float* __restrict__ s_src,
                                                 const float* __restrict__ s_dst,
                                                 float* __restrict__ out) {
  const int lane = threadIdx.x & 31;
  const int e    = blockIdx.x * 8 + (threadIdx.x >> 5);
  if (e >= NEDGES) return;

  const int src = ei[e];
  const int dst = ei[NEDGES + e];

  const float* whd = Wh + (size_t)dst * OUTW;
  __builtin_prefetch(whd + lane, 0, 0);   // global_prefetch while logits compute

  float ev = 0.f;
  if (lane < HEADS) {
    float s = s_src[src * HEADS + lane] + s_dst[dst * HEADS + lane];
    const float* eav = ea + (size_t)e * EDGE_F;
    const float* aeh = a + lane * A_ROW + 2 * OUT_F;
#pragma unroll
    for (int k = 0; k < EDGE_F; ++k) s += eav[k] * aeh[k];
    ev = (s >= 0.f) ? s : ALPHA * s;
  }

  const float e0 = __shfl(ev, 0, 32);
  const float e1 = __shfl(ev, 1, 32);
  const float e2 = __shfl(ev, 2, 32);
  const float e3 = __shfl(ev, 3, 32);
  const float m  = fmaxf(fmaxf(e0, e1), fmaxf(e2, e3));
  const float x0 = expf(e0 - m), x1 = expf(e1 - m);
  const float x2 = expf(e2 - m), x3 = expf(e3 - m);
  const float inv = 1.f / (x0 + x1 + x2 + x3);
  const float att[4] = { x0 * inv, x1 * inv, x2 * inv, x3 * inv };

  float* o = out + (size_t)src * OUTW;
#pragma unroll
  for (int g = 0; g < 4; ++g) {
    const int c = g * 32 + lane;              // head g, coalesced 128B per group
    atomicAdd(&o[c], att[g] * whd[c]);
  }
}

// ---------------------------------------------------------------------------
// Kernel 4: in-place ELU on the accumulated output
// ---------------------------------------------------------------------------
__global__ void gat_elu(float* __restrict__ out) {
  const int i = blockIdx.x * blockDim.x + threadIdx.x;
  if (i < NNODES * OUTW) {
    const float x = out[i];
    out[i] = (x > 0.f) ? x : (expf(x) - 1.f);
  }
}

// ---------------------------------------------------------------------------
extern "C" void kernel_launch(void* const* d_in, const int* in_sizes, int n_in,
                              void* d_out, int out_size, void* d_ws, size_t ws_size,
                              hipStream_t stream) {
  const float* h  = (const float*)d_in[0];   // (N, 128)
  const int*   ei = (const int*)  d_in[1];   // (2, E)
  const float* ea = (const float*)d_in[2];   // (E, 16)
  const float* W  = (const float*)d_in[3];   // (128, 128)
  const float* a  = (const float*)d_in[4];   // (4, 80)
  float* out = (float*)d_out;                // (N, 128) accumulator + final

  float* Wh    = (float*)d_ws;                     // N*128 floats (10.24 MB)
  float* s_src = Wh + (size_t)NNODES * OUTW;       // N*4
  float* s_dst = s_src + (size_t)NNODES * HEADS;   // N*4

  // zero the scatter accumulator every call (deterministic, capture-safe)
  hipMemsetAsync(d_out, 0, sizeof(float) * (size_t)NNODES * OUTW, stream);

  // 1) Wh = h @ W via fp32 WMMA: 1250 row-tiles x 8 col-tiles, 8 waves/block
  gat_gemm_wmma<<<NNODES / 16, 256, 0, stream>>>(h, W, Wh);

  // 2) per-node score tables
  gat_node_scores<<<(NNODES * HEADS + 255) / 256, 256, 0, stream>>>(Wh, a, s_src, s_dst);

  // 3) edge softmax + scatter-add (wave per edge; 640000/8 blocks exactly)
  gat_edges<<<NEDGES / 8, 256, 0, stream>>>(ei, ea, a, Wh, s_src, s_dst, out);

  // 4) ELU epilogue in place
  gat_elu<<<(NNODES * OUTW + 255) / 256, 256, 0, stream>>>(out);
}